// GCN_40140764349028
// MI455X (gfx1250) — compile-verified
//
#include <hip/hip_runtime.h>

// ---------------------------------------------------------------------------
// GCN 2-layer + mean-pool + linear head for MI455X (gfx1250, wave32).
// GEMMs use V_WMMA_F32_16X16X4_F32 (fp32-exact). Edge scatter uses hardware
// fp32 global atomics, one wave (32 lanes == HID) per edge.
// ---------------------------------------------------------------------------

typedef __attribute__((ext_vector_type(2))) float v2f;
typedef __attribute__((ext_vector_type(8))) float v8f;

#define HIDDIM 32

__device__ __forceinline__ void atomicAddF(float* p, float v) {
  unsafeAtomicAdd(p, v);   // lowers to global_atomic_add_f32 on gfx1250
}

// -------------------- utility: zero a float buffer --------------------------
__global__ void gcn_zero_f32(float* __restrict__ p, int n) {
  int i = blockIdx.x * blockDim.x + threadIdx.x;
  if (i < n) p[i] = 0.0f;
}

// -------------------- degree accumulation (in-degree) -----------------------
__global__ void gcn_deg_kernel(const int* __restrict__ dst,
                               float* __restrict__ deg, int nE) {
  int e = blockIdx.x * blockDim.x + threadIdx.x;
  if (e < nE) atomicAddF(&deg[dst[e]], 1.0f);
}

// deg -> dis = rsqrt(deg + 1)  (self-loop folded in; always > 0)
__global__ void gcn_dis_kernel(float* __restrict__ deg, int n) {
  int i = blockIdx.x * blockDim.x + threadIdx.x;
  if (i < n) deg[i] = __frsqrt_rn(deg[i] + 1.0f);
}

// -------------------- WMMA GEMM + fused GCN epilogue ------------------------
// Computes  h = (RELU_IN ? relu(X) : X) @ W        [M,K]x[K,32] -> [M,32]
// Writes    Hs[m]      = h[m] * dis[m]             (edge-gather operand)
//           OutInit[m] = h[m] * dis[m]^2 + bias    (self-loop + bias init)
// One wave per 16-row tile. M must be a multiple of 16 (100000 = 6250*16).
//
// VGPR layouts per CDNA5 ISA 7.12.2:
//   A (16x4 f32): lanes 0-15 hold K={0,1}, lanes 16-31 hold K={2,3}, M = lane%16
//   B (4x16 f32): VGPR0 = rows K0(lanes 0-15)/K2(lanes 16-31), VGPR1 = K1/K3
//   C (16x16 f32): VGPR r -> M = r + 8*(lane/16), N = lane%16
template <bool RELU_IN>
__global__ void gcn_gemm_wmma(const float* __restrict__ X,
                              const float* __restrict__ W,
                              const float* __restrict__ bias,
                              const float* __restrict__ dis,
                              float* __restrict__ Hs,
                              float* __restrict__ OutInit,
                              int M, int K) {
  const int lane = threadIdx.x & 31;
  const int wave = (int)((blockIdx.x * blockDim.x + threadIdx.x) >> 5);
  const int m0 = wave * 16;
  if (m0 >= M) return;
  const int half = lane >> 4;   // 0: K pair {0,1}; 1: K pair {2,3}
  const int l    = lane & 15;

  v8f acc0 = {};   // columns  0..15
  v8f acc1 = {};   // columns 16..31

  const float* xrow = X + (size_t)(m0 + l) * K + 2 * half;

  for (int k = 0; k < K; k += 4) {
    v2f a;
    a.x = xrow[k];
    a.y = xrow[k + 1];
    if (RELU_IN) { a.x = fmaxf(a.x, 0.0f); a.y = fmaxf(a.y, 0.0f); }

    const float* wk = W + (size_t)(k + 2 * half) * HIDDIM;
    v2f b0, b1;
    b0.x = wk[l];               // row (k+2h),   col l
    b0.y = wk[HIDDIM + l];      // row (k+2h+1), col l
    b1.x = wk[16 + l];          // row (k+2h),   col 16+l
    b1.y = wk[HIDDIM + 16 + l]; // row (k+2h+1), col 16+l

    acc0 = __builtin_amdgcn_wmma_f32_16x16x4_f32(false, a, false, b0,
                                                 (short)0, acc0, false, false);
    acc1 = __builtin_amdgcn_wmma_f32_16x16x4_f32(false, a, false, b1,
                                                 (short)0, acc1, false, false);
  }

  const float bLo = bias[l];
  const float bHi = bias[16 + l];
#pragma unroll
  for (int r = 0; r < 8; ++r) {
    const int row = m0 + r + 8 * half;
    const float dr  = dis[row];
    const float hs0 = acc0[r] * dr;
    const float hs1 = acc1[r] * dr;
    Hs[(size_t)row * HIDDIM + l]           = hs0;
    Hs[(size_t)row * HIDDIM + 16 + l]      = hs1;
    OutInit[(size_t)row * HIDDIM + l]      = hs0 * dr + bLo;
    OutInit[(size_t)row * HIDDIM + 16 + l] = hs1 * dr + bHi;
  }
}

// -------------------- edge scatter: out[d] += hs[s] * dis[d] ----------------
// One wave per edge; 32 lanes == 32 features (contiguous 128B row).
__global__ void gcn_edge_agg(const int* __restrict__ src,
                             const int* __restrict__ dst,
                             const float* __restrict__ dis,
                             const float* __restrict__ Hs,
                             float* __restrict__ Out, int nE) {
  const int wave = (int)((blockIdx.x * blockDim.x + threadIdx.x) >> 5);
  const int lane = threadIdx.x & 31;
  if (wave >= nE) return;
  const int s = src[wave];
  const int d = dst[wave];
  const float nd = dis[d];
  const float v  = Hs[(size_t)s * HIDDIM + lane] * nd;
  atomicAddF(&Out[(size_t)d * HIDDIM + lane], v);
}

// -------------------- per-graph mean-pool accumulation ----------------------
__global__ void gcn_pool(const float* __restrict__ Out2,
                         const int* __restrict__ batch,
                         float* __restrict__ sums,
                         float* __restrict__ cnt, int nNodes) {
  const int wave = (int)((blockIdx.x * blockDim.x + threadIdx.x) >> 5);
  const int lane = threadIdx.x & 31;
  if (wave >= nNodes) return;
  const int g = batch[wave];
  atomicAddF(&sums[(size_t)g * HIDDIM + lane], Out2[(size_t)wave * HIDDIM + lane]);
  if (lane == 0) atomicAddF(&cnt[g], 1.0f);
}

// -------------------- final: out[g] = (sums[g]/cnt[g]) . Wl + bl ------------
__global__ void gcn_final(const float* __restrict__ sums,
                          const float* __restrict__ cnt,
                          const float* __restrict__ Wl,
                          const float* __restrict__ bl,
                          float* __restrict__ out, int G) {
  const int wave = (int)((blockIdx.x * blockDim.x + threadIdx.x) >> 5);
  const int lane = threadIdx.x & 31;
  if (wave >= G) return;
  const float c = fmaxf(cnt[wave], 1.0f);
  float v = (sums[(size_t)wave * HIDDIM + lane] / c) * Wl[lane];
#pragma unroll
  for (int off = 16; off > 0; off >>= 1) v += __shfl_xor(v, off, 32);
  if (lane == 0) out[wave] = v + bl[0];
}

// ---------------------------------------------------------------------------
extern "C" void kernel_launch(void* const* d_in, const int* in_sizes, int n_in,
                              void* d_out, int out_size, void* d_ws, size_t ws_size,
                              hipStream_t stream) {
  const float* x     = (const float*)d_in[0];
  const int*   edge  = (const int*)  d_in[1];   // [2, E] row-major: src then dst
  const int*   batch = (const int*)  d_in[2];
  const float* W1    = (const float*)d_in[3];
  const float* b1    = (const float*)d_in[4];
  const float* W2    = (const float*)d_in[5];
  const float* b2    = (const float*)d_in[6];
  const float* Wl    = (const float*)d_in[7];
  const float* bl    = (const float*)d_in[8];
  float* out = (float*)d_out;

  const int N   = in_sizes[2];         // 100000 nodes (len(batch))
  const int E   = in_sizes[1] / 2;     // 3200000 edges
  const int FIN = in_sizes[0] / N;     // 128
  const int G   = out_size;            // 2048 graphs

  const int* srcv = edge;
  const int* dstv = edge + E;

  // Workspace carve-up (~52 MB total as fp32)
  float* ws   = (float*)d_ws;
  float* dis  = ws;                                   // N   (deg, then rsqrt)
  float* Hs1  = dis  + (((size_t)N + 63) & ~(size_t)63);
  float* Out1 = Hs1  + (size_t)N * HIDDIM;
  float* Hs2  = Out1 + (size_t)N * HIDDIM;
  float* Out2 = Hs2  + (size_t)N * HIDDIM;
  float* sums = Out2 + (size_t)N * HIDDIM;            // G*32
  float* cnt  = sums + (size_t)G * HIDDIM;            // G (contiguous w/ sums)

  const int B = 256;  // 8 wave32s per block

  // 1) degrees -> dis
  gcn_zero_f32<<<(N + B - 1) / B, B, 0, stream>>>(dis, N);
  gcn_deg_kernel<<<(E + B - 1) / B, B, 0, stream>>>(dstv, dis, E);
  gcn_dis_kernel<<<(N + B - 1) / B, B, 0, stream>>>(dis, N);

  // zero pooling accumulators (sums + cnt contiguous)
  gcn_zero_f32<<<(G * HIDDIM + G + B - 1) / B, B, 0, stream>>>(sums, G * HIDDIM + G);

  // 2) layer 1: WMMA GEMM (x @ W1) with fused dis-scale + bias/self-loop init
  const int tiles = N / 16;                     // 100000/16 = 6250 waves
  const int gemmBlocks = (tiles * 32 + B - 1) / B;
  gcn_gemm_wmma<false><<<gemmBlocks, B, 0, stream>>>(x, W1, b1, dis, Hs1, Out1, N, FIN);

  // 3) edge scatter layer 1
  const long long eThreads = (long long)E * 32;
  gcn_edge_agg<<<(int)((eThreads + B - 1) / B), B, 0, stream>>>(srcv, dstv, dis, Hs1, Out1, E);

  // 4) layer 2: WMMA GEMM (relu(Out1) @ W2), fused epilogue
  gcn_gemm_wmma<true><<<gemmBlocks, B, 0, stream>>>(Out1, W2, b2, dis, Hs2, Out2, N, HIDDIM);

  // 5) edge scatter layer 2
  gcn_edge_agg<<<(int)((eThreads + B - 1) / B), B, 0, stream>>>(srcv, dstv, dis, Hs2, Out2, E);

  // 6) mean-pool accumulate, then head
  const long long nThreads = (long long)N * 32;
  gcn_pool<<<(int)((nThreads + B - 1) / B), B, 0, stream>>>(Out2, batch, sums, cnt, N);
  gcn_final<<<(G * 32 + B - 1) / B, B, 0, stream>>>(sums, cnt, Wl, bl, out, G);
}